// LocalGrouper_36807869726783
// MI455X (gfx1250) — compile-verified
//
#include <hip/hip_runtime.h>
#include <math.h>

#define BB 16
#define NN 2048
#define DD 128
#define SS 512
#define KK 32
#define EPSF 1e-5f

typedef __attribute__((ext_vector_type(2))) float v2f;
typedef __attribute__((ext_vector_type(8))) float v8f;

// ---------------------------------------------------------------------------
// Kernel 1: furthest point sampling. One 1024-thread block per batch, xyz and
// running min-dist resident in LDS. Per step: per-thread scan (2 pts), wave32
// shfl_xor argmax butterfly (no barrier), 32 partials -> wave0 butterfly.
// Only 2 __syncthreads per sequential step (vs ~11 for an LDS tree), which is
// what matters on the serial critical path. Lowest-index tie-break matches
// jnp.argmax ((v,-i) lexicographic max is associative, butterfly-safe).
// ---------------------------------------------------------------------------
__global__ __launch_bounds__(1024) void fps_kernel(const float* __restrict__ xyz,
                                                   int* __restrict__ fps_idx,
                                                   float* __restrict__ new_xyz) {
  __shared__ float sx[NN], sy[NN], sz[NN];
  __shared__ float dist[NN];
  __shared__ float wv[32];
  __shared__ int   wi[32];
  __shared__ int   sel[SS];
  __shared__ int   sfar;
  const int b = blockIdx.x;
  const int tid = threadIdx.x;
  const int lane = tid & 31;
  const int wid = tid >> 5;
  const float* xb = xyz + (size_t)b * NN * 3;
  for (int n = tid; n < NN; n += 1024) {
    sx[n] = xb[n * 3 + 0];
    sy[n] = xb[n * 3 + 1];
    sz[n] = xb[n * 3 + 2];
    dist[n] = 1e10f;
  }
  __syncthreads();
  int far = 0;
  for (int s = 0; s < SS; ++s) {
    if (tid == 0) sel[s] = far;
    const float cx = sx[far], cy = sy[far], cz = sz[far];
    float bv = -1.0f;
    int bi = 0;
#pragma unroll
    for (int rep = 0; rep < 2; ++rep) {
      const int n = tid + rep * 1024;
      const float dx = sx[n] - cx, dy = sy[n] - cy, dz = sz[n] - cz;
      const float d = dx * dx + dy * dy + dz * dz;
      const float dn = fminf(dist[n], d);
      dist[n] = dn;
      if (dn > bv || (dn == bv && n < bi)) { bv = dn; bi = n; }
    }
    // wave32 argmax butterfly (no barriers)
#pragma unroll
    for (int m = 16; m > 0; m >>= 1) {
      const float ov = __shfl_xor(bv, m, 32);
      const int oi = __shfl_xor(bi, m, 32);
      if (ov > bv || (ov == bv && oi < bi)) { bv = ov; bi = oi; }
    }
    if (lane == 0) { wv[wid] = bv; wi[wid] = bi; }
    __syncthreads();
    if (wid == 0) {
      float v2 = wv[lane];
      int i2 = wi[lane];
#pragma unroll
      for (int m = 16; m > 0; m >>= 1) {
        const float ov = __shfl_xor(v2, m, 32);
        const int oi = __shfl_xor(i2, m, 32);
        if (ov > v2 || (ov == v2 && oi < i2)) { v2 = ov; i2 = oi; }
      }
      if (lane == 0) sfar = i2;
    }
    __syncthreads();
    far = sfar;
  }
  for (int s = tid; s < SS; s += 1024) {
    const int id = sel[s];
    fps_idx[(size_t)b * SS + s] = id;
    new_xyz[((size_t)b * SS + s) * 3 + 0] = sx[id];
    new_xyz[((size_t)b * SS + s) * 3 + 1] = sy[id];
    new_xyz[((size_t)b * SS + s) * 3 + 2] = sz[id];
  }
}

// ---------------------------------------------------------------------------
// Kernel 2: pairwise squared distance via V_WMMA_F32_16X16X4_F32 (unchanged:
// confirmed emitting v_wmma + v_pk_fma_f32 epilogue in round 1).
// ---------------------------------------------------------------------------
__global__ __launch_bounds__(256) void dist_wmma_kernel(const float* __restrict__ xyz,
                                                        const float* __restrict__ new_xyz,
                                                        float* __restrict__ dist) {
  const int stile = blockIdx.x;   // 0..31
  const int b = blockIdx.y;       // 0..15
  const int tid = threadIdx.x;
  const int w = tid >> 5;
  const int lane = tid & 31;
  const bool hi = lane >= 16;
  const int m = lane & 15;
  __shared__ float ssq[16];
  if (tid < 16) {
    const float* nx0 = new_xyz + ((size_t)b * SS + stile * 16 + tid) * 3;
    ssq[tid] = nx0[0] * nx0[0] + nx0[1] * nx0[1] + nx0[2] * nx0[2];
  }
  const float* nx = new_xyz + ((size_t)b * SS + stile * 16 + m) * 3;
  v2f a;
  a.x = hi ? nx[2] : nx[0];
  a.y = hi ? 0.0f : nx[1];
  __syncthreads();
  float ss[8];
#pragma unroll
  for (int r = 0; r < 8; ++r) ss[r] = ssq[r + (hi ? 8 : 0)];
  float* drow = dist + ((size_t)b * SS + (size_t)stile * 16) * NN;
  for (int nt = w; nt < NN / 16; nt += 8) {
    const float* px = xyz + ((size_t)b * NN + nt * 16 + m) * 3;
    const float x0 = px[0], x1 = px[1], x2 = px[2];
    v2f bm;
    bm.x = hi ? x2 : x0;
    bm.y = hi ? 0.0f : x1;
    const float nsq = x0 * x0 + x1 * x1 + x2 * x2;
    v8f acc = {};
    acc = __builtin_amdgcn_wmma_f32_16x16x4_f32(
        /*neg_a=*/false, a, /*neg_b=*/false, bm,
        /*c_mod=*/(short)0, acc, /*reuse_a=*/false, /*reuse_b=*/false);
#pragma unroll
    for (int r = 0; r < 8; ++r) {
      const int row = r + (hi ? 8 : 0);
      drow[(size_t)row * NN + nt * 16 + m] = ss[r] + nsq - 2.0f * acc[r];
    }
  }
}

// ---------------------------------------------------------------------------
// Kernel 3: exact top-K=32 smallest per row. Row is pulled into LDS with
// gfx1250 async-to-LDS b128 transfers (ASYNCcnt-tracked, no VGPR round-trip),
// then 32 selection rounds of wave-shuffle argmin (2 barriers/round).
// ---------------------------------------------------------------------------
__global__ __launch_bounds__(256) void topk_kernel(const float* __restrict__ dist,
                                                   int* __restrict__ knn) {
  const int bs = blockIdx.x;  // b*S + s
  const int tid = threadIdx.x;
  const int lane = tid & 31;
  const int wid = tid >> 5;
  __shared__ float dl[NN];
  __shared__ float wv[8];
  __shared__ int   wi[8];
  const float* drow = dist + (size_t)bs * NN;
  // async global->LDS copy of the row: 256 lanes x 16B x 2 = 8KB
#pragma unroll
  for (int rep = 0; rep < 2; ++rep) {
    const int base = (tid + rep * 256) * 4;
    const unsigned lds_off = (unsigned)(unsigned long long)&dl[base];
    const unsigned long long ga = (unsigned long long)(drow + base);
    asm volatile("global_load_async_to_lds_b128 %0, %1, off"
                 :: "v"(lds_off), "v"(ga)
                 : "memory");
  }
  asm volatile("s_wait_asynccnt 0x0" ::: "memory");
  __syncthreads();
  for (int k = 0; k < KK; ++k) {
    float bv = 3.0e38f;
    int bi = 0;
#pragma unroll
    for (int rep = 0; rep < 8; ++rep) {
      const int n = tid + rep * 256;
      const float v = dl[n];
      if (v < bv || (v == bv && n < bi)) { bv = v; bi = n; }
    }
#pragma unroll
    for (int m = 16; m > 0; m >>= 1) {
      const float ov = __shfl_xor(bv, m, 32);
      const int oi = __shfl_xor(bi, m, 32);
      if (ov < bv || (ov == bv && oi < bi)) { bv = ov; bi = oi; }
    }
    if (lane == 0) { wv[wid] = bv; wi[wid] = bi; }
    __syncthreads();
    if (tid == 0) {
      float v2 = wv[0];
      int i2 = wi[0];
#pragma unroll
      for (int j = 1; j < 8; ++j) {
        const float ov = wv[j];
        const int oi = wi[j];
        if (ov < v2 || (ov == v2 && oi < i2)) { v2 = ov; i2 = oi; }
      }
      knn[(size_t)bs * KK + k] = i2;
      dl[i2] = 3.3e38f;  // remove from future rounds
    }
    __syncthreads();
  }
}

// ---------------------------------------------------------------------------
// Kernel 4: grouped gather + normalization + concat. One wave per (b,s);
// each lane owns 4 consecutive channels so every gather read and every output
// store is b128 (512B/wave contiguous stores on the 256MB-write hot path).
// K=32 float4 values kept in registers. knn indices distributed by __shfl.
// ---------------------------------------------------------------------------
__global__ __launch_bounds__(32) void group_kernel(const float* __restrict__ points,
                                                   const float* __restrict__ alpha,
                                                   const float* __restrict__ beta,
                                                   const int* __restrict__ fps_idx,
                                                   const int* __restrict__ knn,
                                                   float* __restrict__ new_pts) {
  const int bs = blockIdx.x;       // b*SS + s
  const int b = bs >> 9;           // /512
  const int lane = threadIdx.x;    // 0..31
  const int d4 = lane * 4;         // channels d4..d4+3
  const float* pb = points + (size_t)b * NN * DD;
  const int kn_l = knn[(size_t)bs * KK + lane];  // lane k holds index k
  const int fidx = fps_idx[bs];
  const float4 np = *(const float4*)(pb + (size_t)fidx * DD + d4);
  float4 v[KK];
  float s1x = 0.f, s1y = 0.f, s1z = 0.f, s1w = 0.f;
#pragma unroll
  for (int k = 0; k < KK; ++k) {
    const int idx = __shfl(kn_l, k, 32);
    const float4 g = *(const float4*)(pb + (size_t)idx * DD + d4);
    v[k].x = g.x - np.x;
    v[k].y = g.y - np.y;
    v[k].z = g.z - np.z;
    v[k].w = g.w - np.w;
    s1x += v[k].x; s1y += v[k].y; s1z += v[k].z; s1w += v[k].w;
  }
  const float r = 1.0f / (float)KK;
  const float m2x = s1x * r, m2y = s1y * r, m2z = s1z * r, m2w = s1w * r;
  float s2x = 0.f, s2y = 0.f, s2z = 0.f, s2w = 0.f;
#pragma unroll
  for (int k = 0; k < KK; ++k) {
    const float tx = v[k].x - m2x, ty = v[k].y - m2y;
    const float tz = v[k].z - m2z, tw = v[k].w - m2w;
    s2x += tx * tx; s2y += ty * ty; s2z += tz * tz; s2w += tw * tw;
  }
  const float rk = 1.0f / (float)(KK - 1);
  const float ivx = 1.0f / (sqrtf(s2x * rk) + EPSF);
  const float ivy = 1.0f / (sqrtf(s2y * rk) + EPSF);
  const float ivz = 1.0f / (sqrtf(s2z * rk) + EPSF);
  const float ivw = 1.0f / (sqrtf(s2w * rk) + EPSF);
  const float4 al = *(const float4*)(alpha + d4);
  const float4 be = *(const float4*)(beta + d4);
  float4* op4 = (float4*)(new_pts + (size_t)bs * KK * (2 * DD));  // 64 float4 per k
#pragma unroll
  for (int k = 0; k < KK; ++k) {
    float4 g;
    g.x = al.x * (v[k].x * ivx) + be.x;
    g.y = al.y * (v[k].y * ivy) + be.y;
    g.z = al.z * (v[k].z * ivz) + be.z;
    g.w = al.w * (v[k].w * ivw) + be.w;
    op4[(size_t)k * 64 + lane] = g;
    op4[(size_t)k * 64 + 32 + lane] = np;
  }
}

// ---------------------------------------------------------------------------
// d_out layout: [ new_xyz : B*S*3 floats ][ new_pts : B*S*K*256 floats ]
// 64MB distance matrix staged in the new_pts region; consumed by topk before
// group_kernel overwrites it (stream-ordered).
// d_ws: [ fps_idx : B*S ints ][ knn : B*S*K ints ]  (~1.1 MB)
// ---------------------------------------------------------------------------
extern "C" void kernel_launch(void* const* d_in, const int* in_sizes, int n_in,
                              void* d_out, int out_size, void* d_ws, size_t ws_size,
                              hipStream_t stream) {
  const float* xyz    = (const float*)d_in[0];   // [16,2048,3]
  const float* points = (const float*)d_in[1];   // [16,2048,128]
  const float* alpha  = (const float*)d_in[2];   // [128]
  const float* beta   = (const float*)d_in[3];   // [128]

  float* out = (float*)d_out;
  float* new_xyz = out;                          // B*S*3 = 24576 floats
  float* new_pts = out + (size_t)BB * SS * 3;    // B*S*K*256 floats
  float* dist = new_pts;                         // scratch: B*S*N = 16.8M floats

  int* fps_idx = (int*)d_ws;                     // B*S
  int* knn = fps_idx + (size_t)BB * SS;          // B*S*K

  fps_kernel<<<dim3(BB), dim3(1024), 0, stream>>>(xyz, fps_idx, new_xyz);
  dist_wmma_kernel<<<dim3(SS / 16, BB), dim3(256), 0, stream>>>(xyz, new_xyz, dist);
  topk_kernel<<<dim3(BB * SS), dim3(256), 0, stream>>>(dist, knn);
  group_kernel<<<dim3(BB * SS), dim3(32), 0, stream>>>(points, alpha, beta,
                                                       fps_idx, knn, new_pts);
}